// GNNSSM_53300544143410
// MI455X (gfx1250) — compile-verified
//
#include <hip/hip_runtime.h>
#include <hip/hip_bf16.h>
#include <math.h>

// ---------------- problem constants ----------------
constexpr int B      = 64;
constexpr int N      = 1024;
constexpr int FEAT   = 3;
constexpr int HID    = 64;
constexpr int STATE  = 32;
constexpr int NTYPES = 10;
constexpr int TILE   = 64;          // row-block and col-block size
constexpr int NBLK   = N / TILE;    // 16
constexpr int LSTR   = TILE + 1;    // LDS row stride (floats), bank-conflict padding

typedef __attribute__((ext_vector_type(2))) float v2f;
typedef __attribute__((ext_vector_type(8))) float v8f;

__device__ __forceinline__ float sigmoidf_(float x) { return 1.0f / (1.0f + expf(-x)); }

// ============================================================
// Encoder: emb = relu(x @ W1^T + b1) @ W2^T + b2 ; nrm = emb / max(||emb||, 1e-12)
// One wave (32 lanes) per row; each lane owns hidden units (lane, lane+32).
// ============================================================
__global__ void enc_kernel(const float* __restrict__ tf,
                           const float* __restrict__ w1, const float* __restrict__ b1,
                           const float* __restrict__ w2, const float* __restrict__ b2,
                           float* __restrict__ emb, float* __restrict__ nrm) {
    const int wave = threadIdx.x >> 5;
    const int lane = threadIdx.x & 31;
    const int row  = blockIdx.x * 8 + wave;           // 8 waves per block
    const float* x = tf + (size_t)row * FEAT;
    const float x0 = x[0], x1 = x[1], x2 = x[2];

    const int u0 = lane, u1 = lane + 32;
    float h0 = fmaxf(w1[u0*3+0]*x0 + w1[u0*3+1]*x1 + w1[u0*3+2]*x2 + b1[u0], 0.0f);
    float h1 = fmaxf(w1[u1*3+0]*x0 + w1[u1*3+1]*x1 + w1[u1*3+2]*x2 + b1[u1], 0.0f);

    __shared__ float sh[8][HID];
    sh[wave][u0] = h0;
    sh[wave][u1] = h1;
    __syncthreads();

    float e0 = b2[u0], e1 = b2[u1];
    #pragma unroll 8
    for (int d = 0; d < HID; ++d) {
        const float hv = sh[wave][d];
        e0 += hv * w2[u0*HID + d];
        e1 += hv * w2[u1*HID + d];
    }
    // wave-wide sum of squares (wave32)
    float ss = e0*e0 + e1*e1;
    #pragma unroll
    for (int off = 16; off > 0; off >>= 1) ss += __shfl_xor(ss, off, 32);
    const float inv = 1.0f / fmaxf(sqrtf(ss), 1e-12f);

    float* eo = emb + (size_t)row * HID;
    float* no = nrm + (size_t)row * HID;
    eo[u0] = e0; eo[u1] = e1;
    no[u0] = e0 * inv; no[u1] = e1 * inv;
}

// ============================================================
// Fused GNN layer:
//   adj[i,m] = sigmoid(cont[t_i,t_m]) * (nrm_i . nrm_m)
//   hout = relu((hin + adj @ hin) @ W^T + bias)
// Grid: B * NBLK blocks; 256 threads (8 waves). Each wave owns two 16x16
// output subtiles of the 64x64 row block. fp32 WMMA 16x16x4 throughout.
// ============================================================
__global__ void gnn_layer_kernel(const float* __restrict__ nrm,
                                 const int*   __restrict__ types,
                                 const float* __restrict__ contention,
                                 const float* __restrict__ hin,
                                 const float* __restrict__ W,      // [HID][HID]
                                 const float* __restrict__ bias,   // [HID]
                                 float* __restrict__ hout) {
    __shared__ float sNI[TILE * LSTR];   // nrm row-block (resident) ; reused as W in tail
    __shared__ float sN [TILE * LSTR];   // nrm col-block ; reused as adj tile / t tile
    __shared__ float sH [TILE * LSTR];   // h col-block   ; reused as h row-block in tail
    __shared__ float sSig[NTYPES * NTYPES];
    __shared__ int   sTi[TILE];
    __shared__ int   sTm[TILE];

    const int b    = blockIdx.x / NBLK;
    const int i0   = (blockIdx.x % NBLK) * TILE;
    const int tid  = threadIdx.x;
    const int wave = tid >> 5;
    const int lane = tid & 31;

    const int rs = (wave >> 1) * 16;            // subtile row base within 64x64
    const int cA = ((wave & 1) * 2 + 0) * 16;   // subtile col bases
    const int cB = ((wave & 1) * 2 + 1) * 16;
    const int lm = lane & 15;                   // M (A) / N (B,C,D) within subtile
    const int kh = (lane >> 4) * 2;             // per-lane K sub-offset (0 or 2)
    const int rh = (lane >> 4) * 8;             // per-lane C/D row offset (0 or 8)

    // ---- one-time staging ----
    if (tid < NTYPES * NTYPES) sSig[tid] = sigmoidf_(contention[tid]);
    if (tid < TILE)            sTi[tid]  = types[(size_t)b * N + i0 + tid];
    #pragma unroll
    for (int k = 0; k < (TILE*HID)/256; ++k) {
        const int e = tid + k * 256, r = e >> 6, d = e & 63;
        sNI[r * LSTR + d] = nrm[((size_t)b * N + i0 + r) * HID + d];
    }

    v8f acc0 = {}; v8f acc1 = {};

    for (int m0 = 0; m0 < N; m0 += TILE) {
        __syncthreads();   // prior consumers of sN/sH done; init writes visible
        #pragma unroll
        for (int k = 0; k < (TILE*HID)/256; ++k) {
            const int e = tid + k * 256, r = e >> 6, d = e & 63;
            sN[r * LSTR + d] = nrm[((size_t)b * N + m0 + r) * HID + d];
            sH[r * LSTR + d] = hin[((size_t)b * N + m0 + r) * HID + d];
        }
        if (tid < TILE) sTm[tid] = types[(size_t)b * N + m0 + tid];
        __syncthreads();

        // ---- sim tile: nrmI (64x64) @ nrmM^T -> two 16x16 subtiles ----
        v8f s0 = {}; v8f s1 = {};
        #pragma unroll
        for (int kk = 0; kk < HID; kk += 4) {
            v2f a, b0, b1;
            a.x  = sNI[(rs + lm) * LSTR + kk + kh];
            a.y  = sNI[(rs + lm) * LSTR + kk + kh + 1];
            b0.x = sN[(cA + lm) * LSTR + kk + kh];
            b0.y = sN[(cA + lm) * LSTR + kk + kh + 1];
            b1.x = sN[(cB + lm) * LSTR + kk + kh];
            b1.y = sN[(cB + lm) * LSTR + kk + kh + 1];
            s0 = __builtin_amdgcn_wmma_f32_16x16x4_f32(false, a, false, b0, (short)0, s0, false, false);
            s1 = __builtin_amdgcn_wmma_f32_16x16x4_f32(false, a, false, b1, (short)0, s1, false, false);
        }

        __syncthreads();   // all sim reads of sN complete before overwriting with adj
        // ---- adj = sigmoid(contention[ti,tm]) * sim, written D-layout -> LDS ----
        #pragma unroll
        for (int v = 0; v < 8; ++v) {
            const int ri = rs + v + rh;
            const int ti = sTi[ri];
            {
                const int cj = cA + lm;
                sN[ri * LSTR + cj] = sSig[ti * NTYPES + sTm[cj]] * s0[v];
            }
            {
                const int cj = cB + lm;
                sN[ri * LSTR + cj] = sSig[ti * NTYPES + sTm[cj]] * s1[v];
            }
        }
        __syncthreads();

        // ---- agg += adj (64x64) @ hM (64xHID) ----
        #pragma unroll
        for (int kk = 0; kk < TILE; kk += 4) {
            v2f a, b0, b1;
            a.x  = sN[(rs + lm) * LSTR + kk + kh];
            a.y  = sN[(rs + lm) * LSTR + kk + kh + 1];
            b0.x = sH[(kk + kh)     * LSTR + cA + lm];
            b0.y = sH[(kk + kh + 1) * LSTR + cA + lm];
            b1.x = sH[(kk + kh)     * LSTR + cB + lm];
            b1.y = sH[(kk + kh + 1) * LSTR + cB + lm];
            acc0 = __builtin_amdgcn_wmma_f32_16x16x4_f32(false, a, false, b0, (short)0, acc0, false, false);
            acc1 = __builtin_amdgcn_wmma_f32_16x16x4_f32(false, a, false, b1, (short)0, acc1, false, false);
        }
    }

    // ---- tail: hout = relu((hin_I + agg) @ W^T + bias) ----
    __syncthreads();
    #pragma unroll
    for (int k = 0; k < (TILE*HID)/256; ++k) {
        const int e = tid + k * 256, r = e >> 6, d = e & 63;
        sH [r * LSTR + d] = hin[((size_t)b * N + i0 + r) * HID + d];
        sNI[r * LSTR + d] = W[r * HID + d];       // W[o][d]
    }
    __syncthreads();
    #pragma unroll
    for (int v = 0; v < 8; ++v) {
        const int ri = rs + v + rh;
        sN[ri * LSTR + cA + lm] = acc0[v] + sH[ri * LSTR + cA + lm];
        sN[ri * LSTR + cB + lm] = acc1[v] + sH[ri * LSTR + cB + lm];
    }
    __syncthreads();

    v8f o0 = {}; v8f o1 = {};
    #pragma unroll
    for (int kk = 0; kk < HID; kk += 4) {
        v2f a, b0, b1;
        a.x  = sN[(rs + lm) * LSTR + kk + kh];
        a.y  = sN[(rs + lm) * LSTR + kk + kh + 1];
        b0.x = sNI[(cA + lm) * LSTR + kk + kh];      // B[K=d][N=o] = W[o][d]
        b0.y = sNI[(cA + lm) * LSTR + kk + kh + 1];
        b1.x = sNI[(cB + lm) * LSTR + kk + kh];
        b1.y = sNI[(cB + lm) * LSTR + kk + kh + 1];
        o0 = __builtin_amdgcn_wmma_f32_16x16x4_f32(false, a, false, b0, (short)0, o0, false, false);
        o1 = __builtin_amdgcn_wmma_f32_16x16x4_f32(false, a, false, b1, (short)0, o1, false, false);
    }

    #pragma unroll
    for (int v = 0; v < 8; ++v) {
        const int gi = i0 + rs + v + rh;
        {
            const int o = cA + lm;
            hout[((size_t)b * N + gi) * HID + o] = fmaxf(o0[v] + bias[o], 0.0f);
        }
        {
            const int o = cB + lm;
            hout[((size_t)b * N + gi) * HID + o] = fmaxf(o1[v] + bias[o], 0.0f);
        }
    }
}

// ============================================================
// ctx = mean_n h[b,n,:]; GRU cell -> new_state (written to d_out tail)
// One block (96 threads) per batch element.
// ============================================================
__global__ void gru_kernel(const float* __restrict__ h2,
                           const float* __restrict__ prev,
                           const float* __restrict__ w_ih, const float* __restrict__ w_hh,
                           const float* __restrict__ b_ih, const float* __restrict__ b_hh,
                           float* __restrict__ state_out) {
    const int b = blockIdx.x;
    const int t = threadIdx.x;
    __shared__ float ctx[HID];
    __shared__ float gi[3 * STATE];
    __shared__ float gh[3 * STATE];

    if (t < HID) {
        float s = 0.0f;
        const float* p = h2 + (size_t)b * N * HID + t;
        for (int n = 0; n < N; ++n) s += p[(size_t)n * HID];
        ctx[t] = s * (1.0f / (float)N);
    }
    __syncthreads();

    {   // t in [0,96): one gate row each
        float a = b_ih[t];
        #pragma unroll 8
        for (int d = 0; d < HID; ++d) a += ctx[d] * w_ih[t * HID + d];
        float c = b_hh[t];
        #pragma unroll 8
        for (int s = 0; s < STATE; ++s) c += prev[b * STATE + s] * w_hh[t * STATE + s];
        gi[t] = a; gh[t] = c;
    }
    __syncthreads();

    if (t < STATE) {
        const float r  = sigmoidf_(gi[t]           + gh[t]);
        const float z  = sigmoidf_(gi[STATE + t]   + gh[STATE + t]);
        const float nn = tanhf   (gi[2*STATE + t]  + r * gh[2*STATE + t]);
        const float pv = prev[b * STATE + t];
        state_out[b * STATE + t] = (1.0f - z) * nn + z * pv;
    }
}

// ============================================================
// Head: lat = relu([h, state] @ W1^T + b1) @ W2^T + b2
// One wave per row; each lane owns hidden units (lane, lane+32).
// ============================================================
__global__ void head_kernel(const float* __restrict__ h2,
                            const float* __restrict__ state,
                            const float* __restrict__ w1, const float* __restrict__ b1,
                            const float* __restrict__ w2, const float* __restrict__ b2,
                            float* __restrict__ lat) {
    const int wave = threadIdx.x >> 5;
    const int lane = threadIdx.x & 31;
    const int row  = blockIdx.x * 8 + wave;
    const int b    = row / N;
    const float* hp = h2 + (size_t)row * HID;
    const float* sp = state + (size_t)b * STATE;

    const int u0 = lane, u1 = lane + 32;
    float a0 = b1[u0], a1 = b1[u1];
    #pragma unroll 8
    for (int d = 0; d < HID; ++d) {
        const float hv = hp[d];
        a0 += hv * w1[u0 * (HID + STATE) + d];
        a1 += hv * w1[u1 * (HID + STATE) + d];
    }
    #pragma unroll 8
    for (int s = 0; s < STATE; ++s) {
        const float sv = sp[s];
        a0 += sv * w1[u0 * (HID + STATE) + HID + s];
        a1 += sv * w1[u1 * (HID + STATE) + HID + s];
    }
    a0 = fmaxf(a0, 0.0f);
    a1 = fmaxf(a1, 0.0f);
    float acc = a0 * w2[u0] + a1 * w2[u1];
    #pragma unroll
    for (int off = 16; off > 0; off >>= 1) acc += __shfl_xor(acc, off, 32);
    if (lane == 0) lat[row] = acc + b2[0];
}

// ============================================================
extern "C" void kernel_launch(void* const* d_in, const int* in_sizes, int n_in,
                              void* d_out, int out_size, void* d_ws, size_t ws_size,
                              hipStream_t stream) {
    const float* tf       = (const float*)d_in[0];   // (B,N,3)
    const int*   types    = (const int*)  d_in[1];   // (B,N)
    const float* prev     = (const float*)d_in[2];   // (B,32)
    const float* enc_w1   = (const float*)d_in[3];
    const float* enc_b1   = (const float*)d_in[4];
    const float* enc_w2   = (const float*)d_in[5];
    const float* enc_b2   = (const float*)d_in[6];
    const float* cont     = (const float*)d_in[7];   // (10,10)
    const float* gnn_w    = (const float*)d_in[8];   // (2,64,64)
    const float* gnn_b    = (const float*)d_in[9];   // (2,64)
    const float* gru_w_ih = (const float*)d_in[10];
    const float* gru_w_hh = (const float*)d_in[11];
    const float* gru_b_ih = (const float*)d_in[12];
    const float* gru_b_hh = (const float*)d_in[13];
    const float* head_w1  = (const float*)d_in[14];
    const float* head_b1  = (const float*)d_in[15];
    const float* head_w2  = (const float*)d_in[16];
    const float* head_b2  = (const float*)d_in[17];

    float* out_lat   = (float*)d_out;                  // (B,N) = 65536
    float* out_state = (float*)d_out + (size_t)B * N;  // (B,32) = 2048

    // workspace: nrm | hA | hB  (each B*N*HID fp32 = 16 MB)
    float* nrm = (float*)d_ws;
    float* hA  = nrm + (size_t)B * N * HID;
    float* hB  = hA  + (size_t)B * N * HID;

    // 1) encoder -> hA (= emb = h0), nrm
    enc_kernel<<<(B * N) / 8, 256, 0, stream>>>(tf, enc_w1, enc_b1, enc_w2, enc_b2, hA, nrm);

    // 2) two fused GNN layers (adjacency never materialized)
    gnn_layer_kernel<<<B * NBLK, 256, 0, stream>>>(nrm, types, cont, hA,
                                                   gnn_w,              gnn_b,        hB);
    gnn_layer_kernel<<<B * NBLK, 256, 0, stream>>>(nrm, types, cont, hB,
                                                   gnn_w + HID * HID,  gnn_b + HID,  hA);

    // 3) ctx mean + GRU -> new_state (second output)
    gru_kernel<<<B, 96, 0, stream>>>(hA, prev, gru_w_ih, gru_w_hh, gru_b_ih, gru_b_hh, out_state);

    // 4) head -> lat (first output)
    head_kernel<<<(B * N) / 8, 256, 0, stream>>>(hA, out_state, head_w1, head_b1, head_w2, head_b2, out_lat);
}